// RoPE_824633721011
// MI455X (gfx1250) — compile-verified
//
#include <hip/hip_runtime.h>
#include <math.h>

// RoPE for x[B,H,S,D] fp32 with D = 128 (pairwise rotation form).
//
// Roofline: 256MB read + 256MB write @ 23.3 TB/s => ~22us floor. Pure
// bandwidth problem: the whole design is about the memory path.
//
//  Pass 1: build tbl[s][j] = {cos, sin} of pos[s]*theta^(-2j/D)  (2 MB in d_ws)
//  Pass 2: stream x -> out, one wave32 per 128-elem row (lane l -> pairs 2l,2l+1).
//          x/out use NON-TEMPORAL b128 (touch-once, don't evict L2);
//          table loads are regular-temporal (128x reuse, stays hot in L2).
//
// Apply kernel is templated on "S is a power of two" so the hot instantiation
// has a single v_and for the sequence index (no 64-bit modulo expansion).

typedef __attribute__((ext_vector_type(4))) float v4f;

#define LOG2_THETA 13.287712379549449f   // log2(10000)
#define DK 128
#define HALF_DK 64                        // pairs per row
#define VEC_PER_ROW 32                    // float4 per row == one wave32

// ---------------- Pass 1: cos/sin table ----------------
__global__ void rope_build_table(const int* __restrict__ pos,
                                 float2* __restrict__ tbl, int S) {
    int idx = blockIdx.x * blockDim.x + threadIdx.x;
    if (idx >= S * HALF_DK) return;
    int s = idx >> 6;          // idx / 64
    int j = idx & (HALF_DK - 1);
    // inv_freq = theta^(-2j/D) = exp2(-(2j/D) * log2(theta))
    float inv_freq = exp2f(-(float)(2 * j) * (LOG2_THETA / (float)DK));
    float a = (float)pos[s] * inv_freq;
    float sv, cv;
    sincosf(a, &sv, &cv);      // accurate path: a can be up to ~8191 rad
    tbl[idx] = make_float2(cv, sv);
}

// ---------------- Pass 2: streaming apply (table path) ----------------
template <bool POW2>
__global__ void rope_apply(const v4f* __restrict__ x,
                           const v4f* __restrict__ tbl4,   // {c0,s0,c1,s1} per lane
                           v4f* __restrict__ out,
                           unsigned S_or_mask,              // POW2: S-1, else S
                           long long n_vec) {
    long long idx = (long long)blockIdx.x * blockDim.x + threadIdx.x;
    if (idx >= n_vec) return;

    int lane = (int)(idx & (VEC_PER_ROW - 1));      // float4 slot within row
    long long row = idx >> 5;                       // row index (b,h,s flattened)
    int s;
    if (POW2) {
        s = (int)((unsigned)row & S_or_mask);       // single v_and
    } else {
        s = (int)((unsigned)row % S_or_mask);       // 32-bit unsigned mod
    }

    // Table: regular-temporal load -> 2MB table stays resident in 192MB L2,
    // reused 128x (once per (b,h)).
    v4f t = tbl4[(size_t)s * VEC_PER_ROW + lane];   // {cos0, sin0, cos1, sin1}

    // Stream: non-temporal b128 load/store so the 512MB touch-once streams
    // don't evict the table from L2.
    v4f v = __builtin_nontemporal_load(&x[idx]);

    v4f o;
    o.x = t.x * v.x - t.y * v.y;   // cos*even - sin*odd
    o.y = t.y * v.x + t.x * v.y;   // sin*even + cos*odd
    o.z = t.z * v.z - t.w * v.w;
    o.w = t.w * v.z + t.z * v.w;

    __builtin_nontemporal_store(o, &out[idx]);
}

// ---------------- Fallback: inline sincos (only if d_ws < 2MB) ----------------
template <bool POW2>
__global__ void rope_apply_inline(const v4f* __restrict__ x,
                                  const int* __restrict__ pos,
                                  v4f* __restrict__ out,
                                  unsigned S_or_mask,
                                  long long n_vec) {
    long long idx = (long long)blockIdx.x * blockDim.x + threadIdx.x;
    if (idx >= n_vec) return;

    int lane = (int)(idx & (VEC_PER_ROW - 1));
    long long row = idx >> 5;
    int s;
    if (POW2) {
        s = (int)((unsigned)row & S_or_mask);
    } else {
        s = (int)((unsigned)row % S_or_mask);
    }

    int j0 = 2 * lane;                               // first pair index
    float p = (float)pos[s];
    float f0 = exp2f(-(float)(2 * j0) * (LOG2_THETA / (float)DK));
    float f1 = exp2f(-(float)(2 * (j0 + 1)) * (LOG2_THETA / (float)DK));
    float s0, c0, s1, c1;
    sincosf(p * f0, &s0, &c0);
    sincosf(p * f1, &s1, &c1);

    v4f v = __builtin_nontemporal_load(&x[idx]);
    v4f o;
    o.x = c0 * v.x - s0 * v.y;
    o.y = s0 * v.x + c0 * v.y;
    o.z = c1 * v.z - s1 * v.w;
    o.w = s1 * v.z + c1 * v.w;
    __builtin_nontemporal_store(o, &out[idx]);
}

extern "C" void kernel_launch(void* const* d_in, const int* in_sizes, int n_in,
                              void* d_out, int out_size, void* d_ws, size_t ws_size,
                              hipStream_t stream) {
    const float* x   = (const float*)d_in[0];
    const int*   pos = (const int*)d_in[1];
    float*       out = (float*)d_out;

    const int S = in_sizes[1];                       // 4096
    const long long total = (long long)in_sizes[0];  // B*H*S*128
    const long long n_vec = total / 4;               // float4 units
    const bool pow2 = (S & (S - 1)) == 0;
    const unsigned S_or_mask = pow2 ? (unsigned)(S - 1) : (unsigned)S;

    const int threads = 256;                          // 8 wave32 per block
    const unsigned blocks = (unsigned)((n_vec + threads - 1) / threads);

    const size_t tbl_bytes = (size_t)S * HALF_DK * sizeof(float2);  // 2 MB

    if (ws_size >= tbl_bytes) {
        float2* tbl = (float2*)d_ws;
        int tcount = S * HALF_DK;
        rope_build_table<<<(tcount + threads - 1) / threads, threads, 0, stream>>>(
            pos, tbl, S);
        if (pow2) {
            rope_apply<true><<<blocks, threads, 0, stream>>>(
                (const v4f*)x, (const v4f*)tbl, (v4f*)out, S_or_mask, n_vec);
        } else {
            rope_apply<false><<<blocks, threads, 0, stream>>>(
                (const v4f*)x, (const v4f*)tbl, (v4f*)out, S_or_mask, n_vec);
        }
    } else {
        if (pow2) {
            rope_apply_inline<true><<<blocks, threads, 0, stream>>>(
                (const v4f*)x, pos, (v4f*)out, S_or_mask, n_vec);
        } else {
            rope_apply_inline<false><<<blocks, threads, 0, stream>>>(
                (const v4f*)x, pos, (v4f*)out, S_or_mask, n_vec);
        }
    }
}